// EMADechunker_70901320122646
// MI455X (gfx1250) — compile-verified
//
#include <hip/hip_runtime.h>
#include <cstdint>
#include <cstddef>

// Problem constants (match setup_inputs: B=16, J=1024, L=4096, D=512)
#define B_   16
#define J_   1024
#define L_   4096
#define D_   512
#define NG_  8        // groups along J
#define GS_  128      // group size = 8 chunks of 16
#define EPS_ 1e-4f

typedef __attribute__((ext_vector_type(2))) float v2f;
typedef __attribute__((ext_vector_type(4))) float v4f;
typedef __attribute__((ext_vector_type(8))) float v8f;

// ---------------------------------------------------------------------------
// Kernel 1: per-(b,j) coefficients.
//   alpha[b,j]   = mask ? clip(p, EPS, 1-EPS) : 0
//   fppref[b,j]  = prod_{i=g*128..j} (1 - alpha[b,i])   (within-group prefix)
//   fpgroup[b,g] = fppref at group end
// One block per (b,g); 128-thread Hillis-Steele multiply-scan in LDS.
// ---------------------------------------------------------------------------
__global__ void coeff_kernel(const float* __restrict__ conf,
                             const int*   __restrict__ umask,
                             float* __restrict__ alpha,
                             float* __restrict__ fppref,
                             float* __restrict__ fpgroup)
{
    __shared__ float sp[GS_];
    int bg = blockIdx.x;            // b*NG_ + g
    int b  = bg >> 3;
    int g  = bg & 7;
    int t  = threadIdx.x;           // 0..127
    int j  = g * GS_ + t;

    float p = conf[b * J_ + j];
    int   m = umask[b * J_ + j];
    float a = m ? fminf(fmaxf(p, EPS_), 1.0f - EPS_) : 0.0f;
    alpha[b * J_ + j] = a;

    sp[t] = 1.0f - a;
    __syncthreads();
#pragma unroll
    for (int off = 1; off < GS_; off <<= 1) {
        float v = (t >= off) ? sp[t - off] : 1.0f;
        __syncthreads();
        sp[t] *= v;
        __syncthreads();
    }
    fppref[b * J_ + j] = sp[t];
    if (t == GS_ - 1) fpgroup[bg] = sp[t];
}

// ---------------------------------------------------------------------------
// Kernel 2: boundary cumsum -> gather index. One wave32 per batch row,
// ballot-based inclusive prefix count, clipped to [0, J-1].
// ---------------------------------------------------------------------------
__global__ void idx_kernel(const int* __restrict__ bmask, int* __restrict__ idx)
{
    int b    = blockIdx.x;
    int lane = threadIdx.x;         // 0..31
    const int* src = bmask + (size_t)b * L_;
    int*       dst = idx   + (size_t)b * L_;
    int base = 0;
    for (int it = 0; it < L_ / 32; ++it) {
        int l = it * 32 + lane;
        int m = (src[l] != 0) ? 1 : 0;
        unsigned long long bal = __ballot(m);
        int pre = (int)__popcll(bal & ((2ull << lane) - 1ull)); // inclusive
        int v = base + pre - 1;
        v = v < 0 ? 0 : (v > (J_ - 1) ? (J_ - 1) : v);
        dst[l] = v;
        base += (int)__popcll(bal);
    }
}

// ---------------------------------------------------------------------------
// Kernel 3: group-local EMA scan via FP32 WMMA.
// Block = (b, g, dq): 256 threads = 8 waves; wave w owns a 16-column d-tile.
// Per 16-row chunk: build transition T (each lane builds its row of T),
// inject carry as rank-1 C matrix, 4x chained V_WMMA_F32_16X16X4_F32,
// store local smoothed rows, pull next-chunk carry via lane shuffle.
// ---------------------------------------------------------------------------
__global__ void __launch_bounds__(256)
ema_local_kernel(const float* __restrict__ emb,
                 const float* __restrict__ alpha,
                 float* __restrict__ localsm)
{
    __shared__ float sA[GS_];
    int blk = blockIdx.x;           // b*32 + g*4 + dq
    int b   = blk >> 5;
    int g   = (blk >> 2) & 7;
    int dq  = blk & 3;
    int tid = threadIdx.x;

    if (tid < GS_) sA[tid] = alpha[b * J_ + g * GS_ + tid];
    __syncthreads();

    int lane = tid & 31;
    int wave = tid >> 5;
    int col  = lane & 15;           // = M for A rows, = N for B/C/D columns
    int hi   = lane >> 4;           // half-wave select
    int d0   = dq * 128 + wave * 16;

    float state = 0.0f;             // carry (column col), zero at group start

    for (int cc = 0; cc < 8; ++cc) {
        int jbase = g * GS_ + cc * 16;

        float a[16];
#pragma unroll
        for (int i = 0; i < 16; ++i) a[i] = sA[cc * 16 + i];

        // Row (j = col) of T: row[k] = a[k] * prod_{i=k+1..j}(1-a[i]), 0 if k>j
        float row[16];
        {
            float gp = 1.0f;
#pragma unroll
            for (int k = 15; k >= 0; --k) {
                bool act = (k <= col);
                row[k] = act ? a[k] * gp : 0.0f;
                gp     = act ? gp * (1.0f - a[k]) : gp;
            }
        }
        // Within-chunk decay prefix fpc[i] = prod_{t=0..i}(1-a[t])
        float fpc[16];
        {
            float q = 1.0f;
#pragma unroll
            for (int i = 0; i < 16; ++i) { q *= (1.0f - a[i]); fpc[i] = q; }
        }

        // C init = f  (x)  state_in  (rank-1 carry term)
        v8f acc;
#pragma unroll
        for (int r = 0; r < 8; ++r) acc[r] = (hi ? fpc[r + 8] : fpc[r]) * state;

        const float* ebase = emb + ((size_t)b * J_ + jbase) * D_ + d0 + col;
#pragma unroll
        for (int t = 0; t < 4; ++t) {
            v2f A, Bv;
            A[0] = hi ? row[4 * t + 2] : row[4 * t + 0];
            A[1] = hi ? row[4 * t + 3] : row[4 * t + 1];
            int k0 = 4 * t + 2 * hi;
            Bv[0] = __builtin_nontemporal_load(ebase + (size_t)k0 * D_);
            Bv[1] = __builtin_nontemporal_load(ebase + (size_t)(k0 + 1) * D_);
            // D = A(16x4) x B(4x16) + C ; chained over t for K=16
            acc = __builtin_amdgcn_wmma_f32_16x16x4_f32(
                      false, A, false, Bv, (short)0, acc, false, false);
        }

        // Store local smoothed rows (RT: keep resident in 192MB L2 for reuse)
        float* obase = localsm + ((size_t)b * J_ + jbase + 8 * hi) * D_ + d0 + col;
#pragma unroll
        for (int r = 0; r < 8; ++r) obase[(size_t)r * D_] = acc[r];

        // Carry out: smoothed row 15 lives in acc[7] of high half-wave
        state = __shfl(acc[7], 16 + col, 32);

        // Hint-prefetch next chunk's embedding rows (global_prefetch_b8)
        if (cc < 7) {
            const float* pnext = ebase + (size_t)16 * D_;
#pragma unroll
            for (int t = 0; t < 4; ++t)
                __builtin_prefetch(pnext + (size_t)(4 * t + 2 * hi) * D_, 0, 0);
        }
    }
}

// ---------------------------------------------------------------------------
// Kernel 4: group-carry recurrence per (b,d):
//   S_in[b,0,d] = 0 ; S_in[b,g,d] = local_last(g-1) + FPgroup(g-1)*S_in[b,g-1,d]
// ---------------------------------------------------------------------------
__global__ void carry_kernel(const float* __restrict__ localsm,
                             const float* __restrict__ fpgroup,
                             float* __restrict__ sin_)
{
    int gid = blockIdx.x * blockDim.x + threadIdx.x;  // b*D_ + d
    int b = gid >> 9;
    int d = gid & (D_ - 1);
    float S = 0.0f;
#pragma unroll
    for (int g = 0; g < NG_; ++g) {
        sin_[((size_t)b * NG_ + g) * D_ + d] = S;
        float ll = localsm[((size_t)b * J_ + g * GS_ + (GS_ - 1)) * D_ + d];
        float P  = fpgroup[b * NG_ + g];
        S = fmaf(P, S, ll);
    }
}

// ---------------------------------------------------------------------------
// Kernel 5: gather + carry fixup, fused:
//   out[b,l,:] = local[b,j,:] + fppref[b,j] * S_in[b, j>>7, :],  j = idx[b,l]
// local/S_in reads hit L2; 134MB output streamed with NT stores so it does
// not evict the 33.5MB smoothed working set.
// ---------------------------------------------------------------------------
__global__ void gather_kernel(const float* __restrict__ localsm,
                              const float* __restrict__ fppref,
                              const float* __restrict__ sin_,
                              const int*   __restrict__ idx,
                              float* __restrict__ out)
{
    int tid = threadIdx.x;
    int row = blockIdx.x * 2 + (tid >> 7);   // b*L_ + l
    int c   = tid & 127;                     // float4 column
    int b   = row >> 12;                     // L_ = 4096
    int j   = idx[row];
    int g   = j >> 7;                        // GS_ = 128
    float fp = fppref[b * J_ + j];

    const v4f* lp = (const v4f*)localsm;
    const v4f* sp = (const v4f*)sin_;
    v4f lv = lp[((size_t)b * J_  + j) * (D_ / 4) + c];
    v4f sv = sp[((size_t)b * NG_ + g) * (D_ / 4) + c];
    v4f o;
#pragma unroll
    for (int u = 0; u < 4; ++u) o[u] = fmaf(fp, sv[u], lv[u]);
    __builtin_nontemporal_store(o, ((v4f*)out) + (size_t)row * (D_ / 4) + c);
}

// ---------------------------------------------------------------------------
// Host-side launcher
// ---------------------------------------------------------------------------
extern "C" void kernel_launch(void* const* d_in, const int* in_sizes, int n_in,
                              void* d_out, int out_size, void* d_ws, size_t ws_size,
                              hipStream_t stream) {
    (void)in_sizes; (void)n_in; (void)out_size; (void)ws_size;

    const float* emb   = (const float*)d_in[0];  // (B,J,D) f32
    const float* conf  = (const float*)d_in[1];  // (B,J)   f32
    const int*   umask = (const int*)d_in[2];    // (B,J)   bool->int
    const int*   bmask = (const int*)d_in[3];    // (B,L)   bool->int
    float*       out   = (float*)d_out;          // (B,L,D) f32

    // Workspace layout (bytes)
    const size_t SM_B   = (size_t)B_ * J_ * D_ * sizeof(float);   // 33,554,432
    const size_t AL_B   = (size_t)B_ * J_ * sizeof(float);        //     65,536
    const size_t FP_B   = (size_t)B_ * J_ * sizeof(float);        //     65,536
    const size_t FG_B   = (size_t)B_ * NG_ * sizeof(float);       //        512
    const size_t SIN_B  = (size_t)B_ * NG_ * D_ * sizeof(float);  //    262,144

    char* ws = (char*)d_ws;
    float* localsm = (float*)(ws);
    float* alpha   = (float*)(ws + SM_B);
    float* fppref  = (float*)(ws + SM_B + AL_B);
    float* fpgroup = (float*)(ws + SM_B + AL_B + FP_B);
    float* sin_    = (float*)(ws + SM_B + AL_B + FP_B + FG_B);
    int*   idx     = (int*)  (ws + SM_B + AL_B + FP_B + FG_B + SIN_B);

    coeff_kernel    <<<B_ * NG_,          GS_, 0, stream>>>(conf, umask, alpha, fppref, fpgroup);
    idx_kernel      <<<B_,                 32, 0, stream>>>(bmask, idx);
    ema_local_kernel<<<B_ * NG_ * 4,      256, 0, stream>>>(emb, alpha, localsm);
    carry_kernel    <<<(B_ * D_) / 256,   256, 0, stream>>>(localsm, fpgroup, sin_);
    gather_kernel   <<<(B_ * L_) / 2,     256, 0, stream>>>(localsm, fppref, sin_, idx, out);
}